// model_23974507446662
// MI455X (gfx1250) — compile-verified
//
#include <hip/hip_runtime.h>
#include <math.h>

#define NA 4096
#define CUT 5.0f
#define LOG2E 1.44269504088896340736f
#define LN2   0.69314718055994530942f

typedef __attribute__((ext_vector_type(2))) float v2f;
typedef __attribute__((ext_vector_type(8))) float v8f;

__device__ __forceinline__ float frcp(float x)  { return __builtin_amdgcn_rcpf(x); }
__device__ __forceinline__ float fsqrt(float x) { return __builtin_amdgcn_sqrtf(x); }
__device__ __forceinline__ float fexp2(float x) { return __builtin_amdgcn_exp2f(x); }
__device__ __forceinline__ float flog2(float x) { return __builtin_amdgcn_logf(x); }

__device__ __forceinline__ float pow20f(float t) {
    float t2 = t * t;
    float t4 = t2 * t2;
    float t8 = t4 * t4;
    float t16 = t8 * t8;
    return t16 * t4;
}

__global__ void __launch_bounds__(256) eam_init(float* rho, float* out) {
    int t = blockIdx.x * blockDim.x + threadIdx.x;
    if (t < NA) rho[t] = 0.f;
    if (t == 0) out[0] = 0.f;
}

// Row params cached in LDS, pre-transformed:
//   [0]=fe  [1]=beta  [2]=1/re  [3]=lam  [4]=A  [5]=b/fe  [6]=alpha  [7]=kappa
__global__ void __launch_bounds__(256) eam_pair(const float* __restrict__ coords,
                                                const float* __restrict__ params,
                                                float* __restrict__ rho,
                                                float* __restrict__ out) {
    __shared__ float sp[128 * 8];
    __shared__ float snorm[128];

    const int tid  = threadIdx.x;
    const int lane = tid & 31;
    const int wave = tid >> 5;
    const int rowBase = blockIdx.x * 128;

    for (int rl = tid; rl < 128; rl += 256) {
        const float* pr = &params[(rowBase + rl) * 22];
        float re = pr[0], fe = pr[1], al = pr[4], be = pr[5];
        float Aa = pr[6], Bb = pr[7], ka = pr[8], la = pr[9];
        float* s = &sp[rl * 8];
        s[0] = fe;  s[1] = be;  s[2] = frcp(re);     s[3] = la;
        s[4] = Aa;  s[5] = Bb * frcp(fe);  s[6] = al; s[7] = ka;

        float x = coords[(rowBase + rl) * 3 + 0];
        float y = coords[(rowBase + rl) * 3 + 1];
        float z = coords[(rowBase + rl) * 3 + 2];
        snorm[rl] = x * x + y * y + z * z;
    }
    __syncthreads();

    const int waveRow = rowBase + wave * 16;        // this wave's 16 rows
    const int mrow = waveRow + (lane & 15);
    float rx = coords[mrow * 3 + 0];
    float ry = coords[mrow * 3 + 1];
    float rz = coords[mrow * 3 + 2];

    // A (16x4 f32): lanes 0-15 hold K=0,1 ; lanes 16-31 hold K=2,3
    v2f Aop;
    Aop.x = (lane < 16) ? (-2.f * rx) : (-2.f * rz);
    Aop.y = (lane < 16) ? (-2.f * ry) : 1.f;

    // C init = |x_i|^2 per row (VGPR v -> M=v lanes 0-15 ; M=v+8 lanes 16-31)
    v8f Cinit;
    {
        int base = wave * 16 + ((lane < 16) ? 0 : 8);
#pragma unroll
        for (int v = 0; v < 8; ++v) Cinit[v] = snorm[base + v];
    }

    float rho_acc[8];
#pragma unroll
    for (int v = 0; v < 8; ++v) rho_acc[v] = 0.f;
    float pe_acc = 0.f;

    const int colsPerWG = NA / gridDim.y;
    const int colBegin = blockIdx.y * colsPerWG;
    const int colEnd = colBegin + colsPerWG;

    for (int c0 = colBegin; c0 < colEnd; c0 += 16) {
        const int j = c0 + (lane & 15);             // this lane's fixed column
        float jx = coords[j * 3 + 0];
        float jy = coords[j * 3 + 1];
        float jz = coords[j * 3 + 2];
        float jn = jx * jx + jy * jy + jz * jz;

        // B (4x16 f32): lanes 0-15 hold K=0,1 ; lanes 16-31 hold K=2,3
        v2f Bop;
        Bop.x = (lane < 16) ? jx : jz;
        Bop.y = (lane < 16) ? jy : jn;

        // r^2 tile = |x_i|^2 - 2 x_i . x_j + |x_j|^2  via matrix pipe
        v8f r2t = __builtin_amdgcn_wmma_f32_16x16x4_f32(
            false, Aop, false, Bop, (short)0, Cinit, false, false);

        // column params: one fixed j per lane -> registers, hoisted transforms
        const float re_j    = params[j * 22 + 0];
        const float fe_j    = params[j * 22 + 1];
        const float alpha_j = params[j * 22 + 4];
        const float beta_j  = params[j * 22 + 5];
        const float Aa_j    = params[j * 22 + 6];
        const float Bb_j    = params[j * 22 + 7];
        const float kap_j   = params[j * 22 + 8];
        const float lam_j   = params[j * 22 + 9];

        const float ire_j = frcp(re_j);
        const float c0b   = -beta_j * LOG2E;        // exp(-beta(x-1)) = 2^(c0b*x + c1b)
        const float c1b   =  beta_j * LOG2E;
        const float c0a   = -alpha_j * LOG2E;
        const float c1a   =  alpha_j * LOG2E;
        const float bfe_j = Bb_j * frcp(fe_j);

#pragma unroll
        for (int v = 0; v < 8; ++v) {
            const int i = waveRow + v + ((lane < 16) ? 0 : 8);
            float r = fsqrt(fmaxf(r2t[v], 0.f));
            float xj = r * ire_j;
            float ej = fexp2(fmaf(c0b, xj, c1b));
            float frj = fe_j * ej * frcp(1.f + pow20f(xj - lam_j));
            rho_acc[v] += (i != j) ? frj : 0.f;     // rho has NO cutoff; diag excluded
            if (r <= CUT && i != j) {               // ~1% of pairs
                const float* rp = &sp[(wave * 16 + v + ((lane < 16) ? 0 : 8)) * 8];
                float fe_i = rp[0], beta_i = rp[1], ire_i = rp[2], lam_i = rp[3];
                float Aa_i = rp[4], bfe_i = rp[5], alpha_i = rp[6], kap_i = rp[7];
                float xi = r * ire_i;
                float xim1 = xi - 1.f;
                float fri = fe_i * fexp2(-beta_i * LOG2E * xim1) *
                            frcp(1.f + pow20f(xi - lam_i));
                // phi = A e^{-alpha(x-1)}/(1+(x-kappa)^20) - (b/fe) * fr
                float phi_i = Aa_i * fexp2(-alpha_i * LOG2E * xim1) *
                                  frcp(1.f + pow20f(xi - kap_i)) -
                              bfe_i * fri;
                float phi_j = Aa_j * fexp2(fmaf(c0a, xj, c1a)) *
                                  frcp(1.f + pow20f(xj - kap_j)) -
                              bfe_j * frj;
                pe_acc += frj * frcp(fri) * phi_i + fri * frcp(frj) * phi_j;
            }
        }
    }

    // rho: reduce across the 16-lane half that shares a row (xor masks stay in half)
#pragma unroll
    for (int v = 0; v < 8; ++v) {
        float s = rho_acc[v];
        for (int m = 1; m < 16; m <<= 1) s += __shfl_xor(s, m, 32);
        if ((lane & 15) == 0) {
            int i = waveRow + v + ((lane < 16) ? 0 : 8);
            atomicAdd(&rho[i], s);
        }
    }
    // pair energy: full-wave reduce; full-matrix sum * 0.25 == triangle sum * 0.5
    for (int m = 1; m < 32; m <<= 1) pe_acc += __shfl_xor(pe_acc, m, 32);
    if (lane == 0) atomicAdd(out, 0.25f * pe_acc);
}

__global__ void __launch_bounds__(256) eam_embed(const float* __restrict__ params,
                                                 const float* __restrict__ rho,
                                                 float* __restrict__ out) {
    int i = blockIdx.x * blockDim.x + threadIdx.x;
    float F = 0.f;
    if (i < NA) {
        const float* p = &params[i * 22];
        float r = rho[i];
        float rho_e = p[2], rho_s = p[3];
        float eta = p[18], fe2 = p[19], rho_n = p[20], rho_0 = p[21];
        float xn = fmaf(r, frcp(rho_n), -1.f);
        float poly_n = p[10] + xn * (p[11] + xn * (p[12] + xn * p[13]));
        float xe = fmaf(r, frcp(rho_e), -1.f);
        float poly_0 = p[14] + xe * (p[15] + xe * (p[16] + xe * p[17]));
        float l2 = flog2(r * frcp(rho_s));          // log2(rho/rho_s)
        float lnt = LN2 * eta * l2;                 // ln t
        float t = fexp2(eta * l2);                  // t = (rho/rho_s)^eta
        float tail = fe2 * (1.f - lnt) * t;
        F = (r < rho_n) ? poly_n : ((r < rho_0) ? poly_0 : tail);
    }
    for (int m = 1; m < 32; m <<= 1) F += __shfl_xor(F, m, 32);
    if ((threadIdx.x & 31) == 0) atomicAdd(out, F);
}

extern "C" void kernel_launch(void* const* d_in, const int* in_sizes, int n_in,
                              void* d_out, int out_size, void* d_ws, size_t ws_size,
                              hipStream_t stream) {
    const float* coords = (const float*)d_in[0];   // (4096, 3) f32
    const float* params = (const float*)d_in[1];   // (4096, 22) f32
    float* out = (float*)d_out;                    // scalar f32
    float* rho = (float*)d_ws;                     // 4096 f32 scratch

    eam_init<<<dim3((NA + 255) / 256), 256, 0, stream>>>(rho, out);

    dim3 grid(NA / 128, 16);                       // 32 row-blocks x 16 col-splits
    eam_pair<<<grid, 256, 0, stream>>>(coords, params, rho, out);

    eam_embed<<<dim3(NA / 256), 256, 0, stream>>>(params, rho, out);
}